// BinarizeLayer_42958262894616
// MI455X (gfx1250) — compile-verified
//
#include <hip/hip_runtime.h>
#include <cfloat>

// ---------------------------------------------------------------------------
// Otsu binarization for MI455X (gfx1250).
// Memory-bound: ~302 MB compulsory HBM traffic -> ~13us floor at 23.3 TB/s.
// CDNA5 paths: Tensor Data Mover (tensor_load_to_lds, TENSORcnt-pipelined,
// per-wave double buffering) + global_prefetch_b8 on the streaming passes.
// ws layout (u32 words): [0]=min key  [1]=max key  [2..257]=hist  [258]=thr(f32)
// ---------------------------------------------------------------------------

typedef __attribute__((ext_vector_type(4))) unsigned int v4u;
typedef __attribute__((ext_vector_type(8))) int v8i;
typedef __attribute__((ext_vector_type(4))) int v4i;

#define TPB 256
#define WAVES_PER_BLOCK (TPB / 32)
#define W_CHUNK 768            // floats per TDM chunk per wave (divisible by 3)
#define HIST_BLOCKS 1024

#if __has_builtin(__builtin_amdgcn_tensor_load_to_lds)
#define HAS_TDM 1
#else
#define HAS_TDM 0
#endif

// Device-pass-only diagnostics so the compile log proves which path lowered.
#if defined(__AMDGCN__)
#if HAS_TDM
#pragma message("CDNA5 device pass: TDM tensor_load_to_lds path SELECTED")
#else
#pragma message("CDNA5 device pass: TDM builtin NOT available, plain-LDS fallback")
#endif
#endif

__device__ __forceinline__ float grayf(float r, float g, float b) {
  // matches tensordot left-to-right sum with f32 literals of RGB_W
  return (r * 0.2989f + g * 0.5870f) + b * 0.1140f;
}

// monotonic uint encoding of f32 for atomicMin/Max
__device__ __forceinline__ unsigned encf(float f) {
  unsigned u = __float_as_uint(f);
  return (u & 0x80000000u) ? ~u : (u | 0x80000000u);
}
__device__ __forceinline__ float decf(unsigned k) {
  unsigned u = (k & 0x80000000u) ? (k ^ 0x80000000u) : ~k;
  return __uint_as_float(u);
}

// -------------------------- kernel 0: init ws ------------------------------
__global__ void init_ws_kernel(unsigned* __restrict__ ws) {
  int i = threadIdx.x;           // launched with 256 threads
  if (i == 0) { ws[0] = 0xFFFFFFFFu; ws[1] = 0u; ws[258] = 0u; }
  ws[2 + i] = 0u;
}

// ----------------------- kernel 1: gray min/max ----------------------------
__global__ void gray_minmax_kernel(const float* __restrict__ in,
                                   unsigned* __restrict__ ws, long long npix) {
  __shared__ float smin[TPB], smax[TPB];
  const long long quads = npix >> 2;
  const long long gstride = (long long)gridDim.x * blockDim.x;
  const long long gid = (long long)blockIdx.x * blockDim.x + threadIdx.x;
  float lmin = FLT_MAX, lmax = -FLT_MAX;
  for (long long q = gid; q < quads; q += gstride) {
    const float4* p = (const float4*)(in + 12 * q);   // 16B aligned
    __builtin_prefetch(in + 12 * (q + gstride), 0, 0); // global_prefetch_b8
    float4 a = p[0], b = p[1], c = p[2];              // 3x global_load_b128
    float g0 = grayf(a.x, a.y, a.z);
    float g1 = grayf(a.w, b.x, b.y);
    float g2 = grayf(b.z, b.w, c.x);
    float g3 = grayf(c.y, c.z, c.w);
    lmin = fminf(lmin, fminf(fminf(g0, g1), fminf(g2, g3)));
    lmax = fmaxf(lmax, fmaxf(fmaxf(g0, g1), fmaxf(g2, g3)));
  }
  for (long long p = quads * 4 + gid; p < npix; p += gstride) {  // generic tail
    float g = grayf(in[3 * p], in[3 * p + 1], in[3 * p + 2]);
    lmin = fminf(lmin, g); lmax = fmaxf(lmax, g);
  }
  int tid = threadIdx.x;
  smin[tid] = lmin; smax[tid] = lmax;
  __syncthreads();
  for (int s = TPB / 2; s > 0; s >>= 1) {
    if (tid < s) {
      smin[tid] = fminf(smin[tid], smin[tid + s]);
      smax[tid] = fmaxf(smax[tid], smax[tid + s]);
    }
    __syncthreads();
  }
  if (tid == 0) {
    atomicMin(&ws[0], encf(smin[0]));
    atomicMax(&ws[1], encf(smax[0]));
  }
}

// -------------------- TDM helper: 1-D tile -> LDS --------------------------
#if HAS_TDM
__device__ __forceinline__ void tdm_load_1d(unsigned lds_byte_off,
                                            const float* gptr, unsigned nelem) {
  unsigned long long ga = (unsigned long long)(uintptr_t)gptr;
  // force descriptor dwords into SGPRs (values are wave-uniform)
  unsigned w0 = (unsigned)__builtin_amdgcn_readfirstlane((int)1u);  // count=1
  unsigned w1 = (unsigned)__builtin_amdgcn_readfirstlane((int)lds_byte_off);
  unsigned w2 = (unsigned)__builtin_amdgcn_readfirstlane((int)(ga & 0xFFFFFFFFu));
  unsigned w3 = (unsigned)__builtin_amdgcn_readfirstlane(
      (int)(((unsigned)(ga >> 32) & 0x1FFFFFFu) | (2u << 30)));  // type=2 (image)
  v4u g0 = {w0, w1, w2, w3};
  int d0 = __builtin_amdgcn_readfirstlane((int)0x20000u);              // data_size=4B
  int d1 = __builtin_amdgcn_readfirstlane((int)((nelem & 0xFFFFu) << 16)); // tensor_dim0 lo
  int d2 = __builtin_amdgcn_readfirstlane((int)((nelem >> 16) & 0xFFFFu)); // tensor_dim0 hi
  int d3 = __builtin_amdgcn_readfirstlane((int)((nelem & 0xFFFFu) << 16)); // tile_dim0
  v8i g1 = {d0, d1, d2, d3, 0, 0, 0, 0};
  v4i z4 = {0, 0, 0, 0};
#if defined(__clang_major__) && __clang_major__ >= 23
  v8i z8 = {0, 0, 0, 0, 0, 0, 0, 0};
  __builtin_amdgcn_tensor_load_to_lds(g0, g1, z4, z4, z8, 0);
#else
  __builtin_amdgcn_tensor_load_to_lds(g0, g1, z4, z4, 0);
#endif
}
#endif

// --------------------- kernel 2: histogram (TDM) ---------------------------
// dynamic LDS layout: [hist 256*u32 = 1024B][wave0 buf0][wave0 buf1][wave1 buf0]...
__global__ void hist_tdm_kernel(const float* __restrict__ in,
                                unsigned* __restrict__ ws, long long nfloat) {
  extern __shared__ unsigned char smem[];
  unsigned* shHist = (unsigned*)smem;
  float* bufs = (float*)(smem + 1024);

  const int tid = threadIdx.x;
  const int wave = tid >> 5;
  const int lane = tid & 31;

  shHist[tid] = 0u;
  __syncthreads();

  const float vmin = decf(ws[0]);
  const float vmax = decf(ws[1]);
  const float span = (vmax > vmin) ? (vmax - vmin) : 1.0f;

  const long long nchunk = nfloat / W_CHUNK;
  const long long gstride = (long long)gridDim.x * WAVES_PER_BLOCK;
  long long c = (long long)blockIdx.x * WAVES_PER_BLOCK + wave;
  const int npx = W_CHUNK / 3;

#if HAS_TDM
  const unsigned lds_base = __builtin_amdgcn_groupstaticsize();  // dyn-LDS offset
  const unsigned myOff0 = lds_base + 1024u + (unsigned)wave * 2u * W_CHUNK * 4u;
  float* myBuf0 = bufs + (size_t)wave * 2 * W_CHUNK;
  int cur = 0;
  if (c < nchunk) {
    asm volatile("" ::: "memory");
    tdm_load_1d(myOff0, in + c * W_CHUNK, W_CHUNK);
  }
  while (c < nchunk) {
    long long nc = c + gstride;
    bool pre = (nc < nchunk);
    if (pre) {
      asm volatile("" ::: "memory");
      tdm_load_1d(myOff0 + (unsigned)(cur ^ 1) * W_CHUNK * 4u,
                  in + nc * W_CHUNK, W_CHUNK);
    }
    if (pre) __builtin_amdgcn_s_wait_tensorcnt(1);
    else     __builtin_amdgcn_s_wait_tensorcnt(0);
    asm volatile("" ::: "memory");
    const float* B = myBuf0 + (size_t)cur * W_CHUNK;
    for (int k = lane; k < npx; k += 32) {    // stride-3 banks: conflict-free
      float g = grayf(B[3 * k], B[3 * k + 1], B[3 * k + 2]);
      float dn = (g - vmin) / span;
      int idx = (int)floorf(dn * 256.0f);
      idx = idx < 0 ? 0 : (idx > 255 ? 255 : idx);
      atomicAdd(&shHist[idx], 1u);
    }
    c = nc; cur ^= 1;
  }
#else
  // fallback: cooperative wave copy global->LDS, then same LDS processing
  float* myBuf = bufs + (size_t)wave * 2 * W_CHUNK;
  for (; c < nchunk; c += gstride) {
    const float* gp = in + c * W_CHUNK;
    for (int i = lane; i < W_CHUNK; i += 32) myBuf[i] = gp[i];
    for (int k = lane; k < npx; k += 32) {
      float g = grayf(myBuf[3 * k], myBuf[3 * k + 1], myBuf[3 * k + 2]);
      float dn = (g - vmin) / span;
      int idx = (int)floorf(dn * 256.0f);
      idx = idx < 0 ? 0 : (idx > 255 ? 255 : idx);
      atomicAdd(&shHist[idx], 1u);
    }
  }
#endif

  // generic tail (nfloat % W_CHUNK), direct global loads
  {
    long long tailPx0 = (nchunk * W_CHUNK) / 3;
    long long totPx = nfloat / 3;
    long long tstride = (long long)gridDim.x * blockDim.x;
    for (long long p = tailPx0 + (long long)blockIdx.x * blockDim.x + tid;
         p < totPx; p += tstride) {
      float g = grayf(in[3 * p], in[3 * p + 1], in[3 * p + 2]);
      float dn = (g - vmin) / span;
      int idx = (int)floorf(dn * 256.0f);
      idx = idx < 0 ? 0 : (idx > 255 ? 255 : idx);
      atomicAdd(&shHist[idx], 1u);
    }
  }
  __syncthreads();
  unsigned v = shHist[tid];
  if (v) atomicAdd(&ws[2 + tid], v);
}

// ---------------------- kernel 3: Otsu threshold ---------------------------
__global__ void otsu_kernel(unsigned* __restrict__ ws) {
  __shared__ float h[256], hc[256];
  const int tid = threadIdx.x;
  const float vmin = decf(ws[0]);
  const float vmax = decf(ws[1]);
  const float span = (vmax > vmin) ? (vmax - vmin) : 1.0f;
  float cent = vmin + ((float)tid + 0.5f) * span / 256.0f;
  float hv = (float)ws[2 + tid];     // counts < 2^24: exact in f32
  h[tid] = hv;
  hc[tid] = hv * cent;
  __syncthreads();
  if (tid == 0) {
    float total = 0.0f, totsc = 0.0f;
    for (int i = 0; i < 256; ++i) { total += h[i]; totsc += hc[i]; }
    float cw = 0.0f, cs = 0.0f, best = -1.0f;
    int bi = 0;
    for (int t = 0; t < 255; ++t) {
      cw += h[t]; cs += hc[t];
      float w1t = cw;
      float w2t1 = total - cw;                        // w2[t+1]
      float m1 = cs / fmaxf(w1t, 1e-12f);
      float m2 = (totsc - cs) / fmaxf(w2t1, 1e-12f);  // m2[t+1]
      float d = m1 - m2;
      float var = (w1t * w2t1) * (d * d);
      if (var > best) { best = var; bi = t; }         // first max (argmax)
    }
    float thr = vmin + ((float)bi + 0.5f) * span / 256.0f;
    ((float*)ws)[258] = thr;
  }
}

// ----------------------- kernel 4: binarize + tile -------------------------
__global__ void binarize_kernel(const float* __restrict__ in,
                                float* __restrict__ out,
                                const unsigned* __restrict__ ws, long long npix) {
  const float t = ((const float*)ws)[258];
  const long long quads = npix >> 2;
  const long long gstride = (long long)gridDim.x * blockDim.x;
  const long long gid = (long long)blockIdx.x * blockDim.x + threadIdx.x;
  for (long long q = gid; q < quads; q += gstride) {
    const float4* p = (const float4*)(in + 12 * q);
    __builtin_prefetch(in + 12 * (q + gstride), 0, 0); // global_prefetch_b8
    float4 a = p[0], b = p[1], c = p[2];
    float b0 = grayf(a.x, a.y, a.z) > t ? 1.0f : 0.0f;
    float b1 = grayf(a.w, b.x, b.y) > t ? 1.0f : 0.0f;
    float b2 = grayf(b.z, b.w, c.x) > t ? 1.0f : 0.0f;
    float b3 = grayf(c.y, c.z, c.w) > t ? 1.0f : 0.0f;
    float4* o = (float4*)(out + 12 * q);
    o[0] = make_float4(b0, b0, b0, b1);
    o[1] = make_float4(b1, b1, b2, b2);
    o[2] = make_float4(b2, b3, b3, b3);
  }
  for (long long p = quads * 4 + gid; p < npix; p += gstride) {  // tail
    float g = grayf(in[3 * p], in[3 * p + 1], in[3 * p + 2]);
    float v = g > t ? 1.0f : 0.0f;
    out[3 * p] = v; out[3 * p + 1] = v; out[3 * p + 2] = v;
  }
}

// ------------------------------- launcher ----------------------------------
extern "C" void kernel_launch(void* const* d_in, const int* in_sizes, int n_in,
                              void* d_out, int out_size, void* d_ws, size_t ws_size,
                              hipStream_t stream) {
  (void)n_in; (void)out_size; (void)ws_size;
  const float* in = (const float*)d_in[0];
  float* out = (float*)d_out;
  unsigned* ws = (unsigned*)d_ws;

  const long long nflt = (long long)in_sizes[0];
  const long long npix = nflt / 3;
  const long long quads = npix >> 2;
  long long b = (quads + TPB - 1) / TPB;
  if (b < 1) b = 1;
  const int mmBlocks = (int)b;                      // 12288 for the ref shape

  init_ws_kernel<<<1, 256, 0, stream>>>(ws);
  gray_minmax_kernel<<<mmBlocks, TPB, 0, stream>>>(in, ws, npix);

  const size_t smem = 1024 + (size_t)WAVES_PER_BLOCK * 2 * W_CHUNK * 4;  // 50176 B
  long long hb = nflt / W_CHUNK;
  if (hb > HIST_BLOCKS) hb = HIST_BLOCKS;
  if (hb < 1) hb = 1;
  hist_tdm_kernel<<<(int)hb, TPB, smem, stream>>>(in, ws, nflt);

  otsu_kernel<<<1, 256, 0, stream>>>(ws);
  binarize_kernel<<<mmBlocks, TPB, 0, stream>>>(in, out, ws, npix);
}